// Net_17093969838645
// MI455X (gfx1250) — compile-verified
//
#include <hip/hip_runtime.h>

// ---------------------------------------------------------------------------
// SNN (snnTorch Leaky, ATan surrogate fwd = Heaviside) forward on MI455X.
//
// Phase 1: cur1 = x @ W1^T + b1    via V_WMMA_F32_16X16X4_F32 (exact f32)
// Phase 2: 25-step LIF recurrence, one wave32 per batch row, all state
//          (mem1, cur1, W2 rows) register-resident -> single HBM pass.
//
// Roofline: total HBM traffic ~150 MB -> ~6.5us at 23.3 TB/s; GEMM is
// store-bound (65.5 MB out vs 3.3 GFLOP) so f32 WMMA is the right precision.
// cur1 (65.5 MB) fits in the 192 MB L2, so phase 2 likely reads it from L2.
// ---------------------------------------------------------------------------

#define BATCH 16384
#define NI    100
#define NH    1000
#define NO    2
#define NSTEP 25
#define BETA  0.95f
#define THR   1.0f

typedef __attribute__((ext_vector_type(2))) float v2f;
typedef __attribute__((ext_vector_type(8))) float v8f;

// ------------------------- Kernel 1: WMMA GEMM -----------------------------
// One wave computes one 16x16 tile of cur1 [BATCH x NH].
// M tiles = 16384/16 = 1024, N tiles = ceil(1000/16) = 63 (last tile ragged).
// 8 waves / 256-thread block -> 64512/8 = 8064 blocks.
//
// V_WMMA_F32_16X16X4_F32 operand layout (ISA 7.12.2, 32-bit A 16x4):
//   A frag (2 VGPR): lanes 0-15 -> M=lane,    K = k0+0 / k0+1
//                    lanes16-31 -> M=lane-16, K = k0+2 / k0+3
//   B frag (2 VGPR): same striping with N in place of M (B is KxN = W1^T)
//   C/D  (8 VGPR):   VGPR i: lanes 0-15 -> M=i, lanes 16-31 -> M=i+8; N=lane%16
//
// Ragged edge: B rows for n >= 1000 are clamped to row 999. The resulting
// garbage only lands in D columns n >= 1000, and each lane's accumulators all
// belong to its own column (N = lane%16) -- exactly the lanes whose stores are
// guarded off. Clamped values are finite W1 entries, so no NaN propagation.
// => no masking needed in the K-loop; loads feed WMMAs directly.
__global__ __launch_bounds__(256) void fc1_wmma(const float* __restrict__ x,
                                                const float* __restrict__ W1,
                                                const float* __restrict__ b1,
                                                float* __restrict__ cur1)
{
    const int lane   = threadIdx.x & 31;
    const int waveId = blockIdx.x * 8 + (threadIdx.x >> 5);
    const int NT_N   = 63;
    const int tileM  = waveId / NT_N;
    const int tileN  = waveId % NT_N;
    const int row0   = tileM * 16;
    const int col0   = tileN * 16;

    const int half = lane >> 4;        // 0: lanes 0-15, 1: lanes 16-31
    const int idx  = lane & 15;        // M index for A, N index for B
    const int koff = half * 2;         // K sub-offset inside the 4-slice

    const int arow = row0 + idx;                     // always < BATCH
    const int ncol = col0 + idx;                     // may exceed NH-1
    const int ncl  = (ncol < NH) ? ncol : (NH - 1);  // clamped, keeps EXEC uniform

    const float* aptr = x  + (size_t)arow * NI + koff; // 8B aligned (NI even, koff even)
    const float* bptr = W1 + (size_t)ncl  * NI + koff; // B[k][n] = W1[n][k]

    v8f acc = {};
#pragma unroll
    for (int k0 = 0; k0 < NI; k0 += 4) {               // 25 WMMAs
        v2f a = *reinterpret_cast<const v2f*>(aptr + k0);
        v2f b = *reinterpret_cast<const v2f*>(bptr + k0);
        acc = __builtin_amdgcn_wmma_f32_16x16x4_f32(
            /*neg_a=*/false, a, /*neg_b=*/false, b,
            /*c_mod=*/(short)0, acc, /*reuse_a=*/false, /*reuse_b=*/false);
    }

    const float bias = b1[ncl];                        // per-column bias
    if (ncol < NH) {
#pragma unroll
        for (int i = 0; i < 8; ++i) {
            const int r = row0 + i + half * 8;
            cur1[(size_t)r * NH + ncol] = acc[i] + bias;
        }
    }
}

// --------------------- Kernel 2: register-resident LIF ---------------------
// One wave32 per batch row. Lane l owns h = 32*i + l, i = 0..31 (1024 slots,
// 1000 valid; pads have cur=w2=0 so they never spike / never contribute).
// mem1 (32), cur1 (32), W2 row0 (32), W2 row1 (32) live in VGPRs: the 25-step
// loop touches no memory except 2 float2 outputs per step from lane 0.
__global__ __launch_bounds__(256) void snn_steps(const float* __restrict__ cur1,
                                                 const float* __restrict__ W2,
                                                 const float* __restrict__ b2,
                                                 float* __restrict__ out)
{
    const int lane = threadIdx.x & 31;
    const int row  = blockIdx.x * 8 + (threadIdx.x >> 5);   // batch index

    float cur[32], mem[32], w20[32], w21[32];
    const float* cbase = cur1 + (size_t)row * NH;
#pragma unroll
    for (int i = 0; i < 32; ++i) {
        const int  h  = i * 32 + lane;                      // coalesced per i
        const bool v  = (h < NH);
        const int  hc = v ? h : 0;
        const float m = v ? 1.0f : 0.0f;
        cur[i] = cbase[hc] * m;
        w20[i] = W2[hc] * m;          // W2[0, h]
        w21[i] = W2[NH + hc] * m;     // W2[1, h]
        mem[i] = 0.0f;
    }

    const float b20 = b2[0], b21 = b2[1];
    float mem2_0 = 0.0f, mem2_1 = 0.0f;
    const size_t recOff = (size_t)NSTEP * BATCH * NO;       // spk2_rec section

    for (int t = 0; t < NSTEP; ++t) {
        float s0 = 0.0f, s1 = 0.0f;
#pragma unroll
        for (int i = 0; i < 32; ++i) {
            float m = mem[i];
            const float reset = (m > THR) ? THR : 0.0f;     // from OLD mem (detached)
            m = BETA * m + cur[i] - reset;
            mem[i] = m;
            const float spk = (m > THR) ? 1.0f : 0.0f;
            s0 = fmaf(spk, w20[i], s0);
            s1 = fmaf(spk, w21[i], s1);
        }
        // wave32 butterfly: every lane ends with the full 1000-term dot
#pragma unroll
        for (int off = 16; off > 0; off >>= 1) {
            s0 += __shfl_xor(s0, off, 32);
            s1 += __shfl_xor(s1, off, 32);
        }
        const float c0 = s0 + b20, c1 = s1 + b21;
        const float r0 = (mem2_0 > THR) ? THR : 0.0f;
        const float r1 = (mem2_1 > THR) ? THR : 0.0f;
        mem2_0 = BETA * mem2_0 + c0 - r0;
        mem2_1 = BETA * mem2_1 + c1 - r1;
        if (lane == 0) {
            const size_t o = ((size_t)t * BATCH + row) * NO;   // even index -> 8B aligned
            *reinterpret_cast<float2*>(out + o) =
                make_float2(mem2_0, mem2_1);
            *reinterpret_cast<float2*>(out + recOff + o) =
                make_float2((mem2_0 > THR) ? 1.0f : 0.0f,
                            (mem2_1 > THR) ? 1.0f : 0.0f);
        }
    }
}

// ------------------------------- Launcher ----------------------------------
extern "C" void kernel_launch(void* const* d_in, const int* in_sizes, int n_in,
                              void* d_out, int out_size, void* d_ws, size_t ws_size,
                              hipStream_t stream)
{
    const float* x  = (const float*)d_in[0];   // [16384, 100]
    const float* W1 = (const float*)d_in[1];   // [1000, 100]
    const float* b1 = (const float*)d_in[2];   // [1000]
    const float* W2 = (const float*)d_in[3];   // [2, 1000]
    const float* b2 = (const float*)d_in[4];   // [2]
    float* out  = (float*)d_out;               // [25,16384,2] mem2 ++ [25,16384,2] spk2
    float* cur1 = (float*)d_ws;                // [16384, 1000] staging (65.5 MB)

    // 1024 M-tiles * 63 N-tiles = 64512 waves, 8 waves/block
    fc1_wmma<<<8064, 256, 0, stream>>>(x, W1, b1, cur1);
    // 16384 rows, one wave32 per row, 8 rows/block
    snn_steps<<<2048, 256, 0, stream>>>(cur1, W2, b2, out);
}